// SimpleProbabilityModel_23699629540184
// MI455X (gfx1250) — compile-verified
//
#include <hip/hip_runtime.h>
#include <math.h>

// ---------------------------------------------------------------------------
// SoftLength + Normal-CDF, MI455X (gfx1250).
// x: [N=8192, B=4096] f32 row-major.  out: [B] f32.
// Segmented suffix-cummax (sigmoid is monotone => cummax on raw x),
// streamed through LDS with gfx1250 async global->LDS copies.
// ---------------------------------------------------------------------------

#define SEG   128                 // row segments (N/SEG = 64 rows each)
#define CHUNK 8                   // rows per async-staged LDS chunk
#define TPB   256                 // 8 wave32s
#define COLS_PER_BLOCK (TPB * 4)  // float4 per thread -> 1024 columns/block

typedef int v4i __attribute__((ext_vector_type(4)));
typedef __attribute__((address_space(1))) v4i gbl_v4i_t;
typedef __attribute__((address_space(3))) v4i lds_v4i_t;

// One async copy per lane of 16 bytes from the global address gsrc to the LDS address ldst. Tracked by ASYNCcnt.
__device__ __forceinline__ void async_copy_b128(const float* gsrc, void* ldst) {
  __builtin_amdgcn_global_load_async_to_lds_b128(
      (gbl_v4i_t*)gsrc, (lds_v4i_t*)ldst, /*offset=*/0, /*cpol=*/0);
}

template <int N>
__device__ __forceinline__ void wait_async() {
#if __has_builtin(__builtin_amdgcn_s_wait_asynccnt)
  __builtin_amdgcn_s_wait_asynccnt(N);
#else
  asm volatile("s_wait_asynccnt %0" ::"n"(N) : "memory");
#endif
  asm volatile("" ::: "memory");  // keep LDS reads below the wait
}

__device__ __forceinline__ float sigmoidf_fast(float v) {
  return 1.0f / (1.0f + __expf(-v));
}

// Normal(0.5*S, sqrt(0.01*S)).cdf(4096) = 0.5*(1+erf((4096-0.5S)/sqrt(0.02S)))
__device__ __forceinline__ float prob_from_len(float S) {
  const float z = (4096.0f - 0.5f * S) / sqrtf(0.02f * S);
  return 0.5f * (1.0f + erff(z));
}

// K1: segmax[seg][b] = max over the segment's rows of raw x[.,b]
__global__ void __launch_bounds__(TPB)
k_segmax(const float* __restrict__ x, float* __restrict__ segmax,
         int B, int rowsPerSeg) {
  __shared__ float4 tile[2][CHUNK][TPB];  // 64 KB
  const int tid = threadIdx.x;
  const int seg = blockIdx.y;
  const int c0  = blockIdx.x * COLS_PER_BLOCK + tid * 4;
  const long rowBase = (long)seg * rowsPerSeg;
  const int nch = rowsPerSeg / CHUNK;

  for (int r = 0; r < CHUNK; ++r)
    async_copy_b128(x + (rowBase + r) * (long)B + c0, &tile[0][r][tid]);

  float4 m = make_float4(-INFINITY, -INFINITY, -INFINITY, -INFINITY);
  for (int k = 0; k < nch; ++k) {
    const int cur = k & 1;
    if (k + 1 < nch) {
      const long rb = rowBase + (long)(k + 1) * CHUNK;
      for (int r = 0; r < CHUNK; ++r)
        async_copy_b128(x + (rb + r) * (long)B + c0, &tile[cur ^ 1][r][tid]);
      wait_async<CHUNK>();   // chunk k fully landed (cnt <= 8 outstanding)
    } else {
      wait_async<0>();
    }
    for (int r = 0; r < CHUNK; ++r) {
      float4 v = tile[cur][r][tid];
      m.x = fmaxf(m.x, v.x);
      m.y = fmaxf(m.y, v.y);
      m.z = fmaxf(m.z, v.z);
      m.w = fmaxf(m.w, v.w);
    }
  }
  *(float4*)(segmax + (long)seg * B + c0) = m;
}

// K2: in-place exclusive suffix max over segments (per column).
__global__ void k_excl_suffix_max(float* __restrict__ buf, int B) {
  const int b = blockIdx.x * blockDim.x + threadIdx.x;
  if (b >= B) return;
  float run = -INFINITY;
  for (int s = SEG - 1; s >= 0; --s) {
    const long idx = (long)s * B + b;
    const float v = buf[idx];
    buf[idx] = run;           // exclusive: max over segments AFTER s
    run = fmaxf(run, v);
  }
}

// K3: per-segment partial sums of sigmoid(suffix running max), rows descending.
__global__ void __launch_bounds__(TPB)
k_suffix_partial(const float* __restrict__ x, const float* __restrict__ mafter,
                 float* __restrict__ partial, int B, int rowsPerSeg) {
  __shared__ float4 tile[2][CHUNK][TPB];  // 64 KB
  const int tid = threadIdx.x;
  const int seg = blockIdx.y;
  const int c0  = blockIdx.x * COLS_PER_BLOCK + tid * 4;
  const long rowBase = (long)seg * rowsPerSeg;
  const int nch = rowsPerSeg / CHUNK;

  {
    const int k = nch - 1;
    const long rb = rowBase + (long)k * CHUNK;
    for (int r = 0; r < CHUNK; ++r)
      async_copy_b128(x + (rb + r) * (long)B + c0, &tile[k & 1][r][tid]);
  }

  float4 run = *(const float4*)(mafter + (long)seg * B + c0);
  float4 acc = make_float4(0.f, 0.f, 0.f, 0.f);

  for (int k = nch - 1; k >= 0; --k) {
    const int cur = k & 1;
    if (k > 0) {
      const long rb = rowBase + (long)(k - 1) * CHUNK;
      for (int r = 0; r < CHUNK; ++r)
        async_copy_b128(x + (rb + r) * (long)B + c0, &tile[cur ^ 1][r][tid]);
      wait_async<CHUNK>();
    } else {
      wait_async<0>();
    }
    for (int r = CHUNK - 1; r >= 0; --r) {
      float4 v = tile[cur][r][tid];
      run.x = fmaxf(run.x, v.x); acc.x += sigmoidf_fast(run.x);
      run.y = fmaxf(run.y, v.y); acc.y += sigmoidf_fast(run.y);
      run.z = fmaxf(run.z, v.z); acc.z += sigmoidf_fast(run.z);
      run.w = fmaxf(run.w, v.w); acc.w += sigmoidf_fast(run.w);
    }
  }
  *(float4*)(partial + (long)seg * B + c0) = acc;
}

// K4: reduce 128 partials per column, apply Normal CDF.
__global__ void k_finalize(const float* __restrict__ partial,
                           float* __restrict__ out, int B) {
  const int c0 = (blockIdx.x * blockDim.x + threadIdx.x) * 4;
  if (c0 >= B) return;
  float4 s = make_float4(0.f, 0.f, 0.f, 0.f);
  for (int sg = 0; sg < SEG; ++sg) {
    float4 p = *(const float4*)(partial + (long)sg * B + c0);
    s.x += p.x; s.y += p.y; s.z += p.z; s.w += p.w;
  }
  float4 o;
  o.x = prob_from_len(s.x);
  o.y = prob_from_len(s.y);
  o.z = prob_from_len(s.z);
  o.w = prob_from_len(s.w);
  *(float4*)(out + c0) = o;
}

extern "C" void kernel_launch(void* const* d_in, const int* in_sizes, int n_in,
                              void* d_out, int out_size, void* d_ws, size_t ws_size,
                              hipStream_t stream) {
  const float* x = (const float*)d_in[0];
  float* out = (float*)d_out;
  const int B = out_size;             // 4096
  const int N = in_sizes[0] / B;      // 8192
  const int rowsPerSeg = N / SEG;     // 64 (multiple of CHUNK)

  float* segmax  = (float*)d_ws;                 // SEG*B f32 (2 MB)
  float* partial = segmax + (size_t)SEG * B;     // SEG*B f32 (2 MB)

  dim3 gridA(B / COLS_PER_BLOCK, SEG);           // (4, 128)
  k_segmax<<<gridA, TPB, 0, stream>>>(x, segmax, B, rowsPerSeg);
  k_excl_suffix_max<<<(B + 255) / 256, 256, 0, stream>>>(segmax, B);
  k_suffix_partial<<<gridA, TPB, 0, stream>>>(x, segmax, partial, B, rowsPerSeg);
  k_finalize<<<(B / 4 + 255) / 256, 256, 0, stream>>>(partial, out, B);
}